// GadgetTwoMLPPredictor_58188216926849
// MI455X (gfx1250) — compile-verified
//
#include <hip/hip_runtime.h>
#include <hip/hip_bf16.h>

#define B_   64
#define S_   512
#define Z_   1024
#define H1_  512
#define H2_  256
#define N_   (Z_ * S_)        // 524288 = 2^19
#define ZCH  8
#define ZCHLEN (Z_ / ZCH)     // 128

typedef __attribute__((ext_vector_type(2))) float v2f;
typedef __attribute__((ext_vector_type(8))) float v8f;

#define NEG_INF (-__builtin_inff())

// ---------------- log-sum-exp helpers (wave32) ----------------

__device__ __forceinline__ void lse_acc(float& m, float& s, float x) {
  if (x > m) { s = s * expf(m - x) + 1.0f; m = x; }
  else       { s += expf(x - m); }
}

__device__ __forceinline__ void lse_merge(float& m, float& s, float mo, float so) {
  float M = fmaxf(m, mo);
  if (M == NEG_INF) { m = M; s = 0.0f; return; }
  s = s * expf(m - M) + so * expf(mo - M);
  m = M;
}

__device__ __forceinline__ float wave_lse(float m, float s) {
  #pragma unroll
  for (int off = 16; off > 0; off >>= 1) {
    float mo = __shfl_xor(m, off, 32);
    float so = __shfl_xor(s, off, 32);
    lse_merge(m, s, mo, so);
  }
  return (s > 0.0f) ? m + logf(s) : NEG_INF;
}

// ---------------- small prologue kernels ----------------

__global__ void k_zlogits(const float* __restrict__ prior, float* __restrict__ zlog) {
  __shared__ float red[256];
  int t = threadIdx.x;
  float mx = NEG_INF;
  for (int i = t; i < Z_; i += 256) mx = fmaxf(mx, prior[i]);
  red[t] = mx; __syncthreads();
  for (int o = 128; o; o >>= 1) { if (t < o) red[t] = fmaxf(red[t], red[t + o]); __syncthreads(); }
  float g = red[0]; __syncthreads();
  float sm = 0.0f;
  for (int i = t; i < Z_; i += 256) sm += expf(prior[i] - g);
  red[t] = sm; __syncthreads();
  for (int o = 128; o; o >>= 1) { if (t < o) red[t] += red[t + o]; __syncthreads(); }
  float lse = g + logf(red[0]);
  for (int i = t; i < Z_; i += 256) zlog[i] = prior[i] - lse;
}

__global__ void k_softmax(const float* __restrict__ sl, float* __restrict__ p) {
  __shared__ float red[256];
  int b = blockIdx.x, t = threadIdx.x;
  const float* row = sl + b * S_;
  float x0 = row[t], x1 = row[t + 256];
  red[t] = fmaxf(x0, x1); __syncthreads();
  for (int o = 128; o; o >>= 1) { if (t < o) red[t] = fmaxf(red[t], red[t + o]); __syncthreads(); }
  float g = red[0]; __syncthreads();
  float e0 = expf(x0 - g), e1 = expf(x1 - g);
  red[t] = e0 + e1; __syncthreads();
  for (int o = 128; o; o >>= 1) { if (t < o) red[t] += red[t + o]; __syncthreads(); }
  float inv = 1.0f / red[0];
  p[b * S_ + t]       = e0 * inv;
  p[b * S_ + t + 256] = e1 * inv;
}

// out[b,j] = sigmoid(sum_k in[b,k]*W[k,j] + bias[j])
__global__ void k_dense_sigmoid(const float* __restrict__ in, const float* __restrict__ W,
                                const float* __restrict__ bias, float* __restrict__ out,
                                int K, int N) {
  int nblk = N >> 8;
  int b = blockIdx.x / nblk;
  int j = (blockIdx.x % nblk) * 256 + threadIdx.x;
  const float* inr = in + b * K;
  float acc = bias[j];
  for (int k = 0; k < K; ++k) acc = fmaf(inr[k], W[(size_t)k * N + j], acc);
  out[b * N + j] = 1.0f / (1.0f + expf(-acc));
}

__global__ void k_zero(float* __restrict__ p, int n) {
  int i = blockIdx.x * 256 + threadIdx.x;
  if (i < n) p[i] = 0.0f;
}

// ---------------- WMMA fp32 GEMM: lc0[64, N_] = V[64,256] x Wout[256, N_] + bout ----------------

__global__ void __launch_bounds__(256) k_gemm_wmma(const float* __restrict__ V,
                                                   const float* __restrict__ Wout,
                                                   const float* __restrict__ bout,
                                                   float* __restrict__ out) {
  __shared__ float ldsA[H2_ * 64];   // [k][m], 64 KB
  int tid = threadIdx.x;
  for (int idx = tid; idx < 64 * H2_; idx += 256) {
    int m = idx >> 8;      // 0..63
    int k = idx & 255;     // coalesced global reads
    ldsA[k * 64 + m] = V[m * H2_ + k];
  }
  __syncthreads();

  int wave  = tid >> 5;
  int lane  = tid & 31;
  int lhalf = lane >> 4;
  int l15   = lane & 15;
  int ncol  = blockIdx.x * 128 + wave * 16 + l15;

  v8f acc0 = {}, acc1 = {}, acc2 = {}, acc3 = {};
  const float* wp = Wout + (size_t)(lhalf * 2) * N_ + ncol;
  int krow = lhalf * 2;

  for (int kk = 0; kk < 64; ++kk) {
    v2f bf;
    bf.x = wp[0];
    bf.y = wp[(size_t)N_];
    const float* ap = &ldsA[krow * 64 + l15];
    v2f a0; a0.x = ap[0];  a0.y = ap[64];
    v2f a1; a1.x = ap[16]; a1.y = ap[80];
    v2f a2; a2.x = ap[32]; a2.y = ap[96];
    v2f a3; a3.x = ap[48]; a3.y = ap[112];
    acc0 = __builtin_amdgcn_wmma_f32_16x16x4_f32(false, a0, false, bf, (short)0, acc0, false, false);
    acc1 = __builtin_amdgcn_wmma_f32_16x16x4_f32(false, a1, false, bf, (short)0, acc1, false, false);
    acc2 = __builtin_amdgcn_wmma_f32_16x16x4_f32(false, a2, false, bf, (short)0, acc2, false, false);
    acc3 = __builtin_amdgcn_wmma_f32_16x16x4_f32(false, a3, false, bf, (short)0, acc3, false, false);
    wp += (size_t)4 * N_;
    krow += 4;
  }

  float bv = bout[ncol];
  float* op = out + ncol;
  #pragma unroll
  for (int r = 0; r < 8; ++r) {
    int mrow = r + 8 * lhalf;
    op[(size_t)(mrow)      * N_] = acc0[r] + bv;
    op[(size_t)(16 + mrow) * N_] = acc1[r] + bv;
    op[(size_t)(32 + mrow) * N_] = acc2[r] + bv;
    op[(size_t)(48 + mrow) * N_] = acc3[r] + bv;
  }
}

// ---------------- Sinkhorn / rejection reductions over fixed lc0 ----------------

// mode 0: u[b,z] = lp1[z] - lse_s(lc0[b,z,:] + t[b,:])
// mode 1: A2[b,z] = u11[b,z]-lp1[z]; R[b,z] from log_keep = A2 + lse_s(lc0 + Tacc)
__global__ void k_row(const float* __restrict__ lc0, const float* __restrict__ tcol,
                      const float* __restrict__ lp1, const float* __restrict__ uin,
                      float* __restrict__ out1, float* __restrict__ out2, int mode) {
  int row  = blockIdx.x * 8 + (threadIdx.x >> 5);
  int lane = threadIdx.x & 31;
  int b = row >> 10;
  int z = row & (Z_ - 1);
  const float4* lr4 = (const float4*)(lc0 + (size_t)row * S_);
  const float4* tr4 = (const float4*)(tcol + b * S_);
  float m = NEG_INF, s = 0.0f;
  #pragma unroll
  for (int it = 0; it < 4; ++it) {
    int i4 = it * 32 + lane;                 // float4 index within the 512-row
    float4 a  = lr4[i4];
    float4 tv = tr4[i4];
    lse_acc(m, s, a.x + tv.x);
    lse_acc(m, s, a.y + tv.y);
    lse_acc(m, s, a.z + tv.z);
    lse_acc(m, s, a.w + tv.w);
  }
  float lse = wave_lse(m, s);
  if (lane == 0) {
    if (mode == 0) {
      out1[row] = lp1[z] - lse;
    } else {
      float A  = uin[row] - lp1[z];
      float lk = A + lse;                    // log_keep
      float ek = expf(lk);
      out1[row] = A;
      out2[row] = (ek >= 1.0f) ? NEG_INF : log1pf(-ek);
    }
  }
}

// partial lse over a z-chunk: 128 threads x 4 consecutive columns, float4 loads
__global__ void k_col_partial(const float* __restrict__ lc0, const float* __restrict__ u,
                              float* __restrict__ pm, float* __restrict__ ps) {
  int b  = blockIdx.x / ZCH;
  int ch = blockIdx.x % ZCH;
  int tid = threadIdx.x;                     // 128 threads
  const float4* base = (const float4*)(lc0 + ((size_t)b * Z_ + (size_t)ch * ZCHLEN) * S_) + tid;
  const float* ub = u + b * Z_ + ch * ZCHLEN;
  float m0 = NEG_INF, s0 = 0.f, m1 = NEG_INF, s1 = 0.f;
  float m2 = NEG_INF, s2 = 0.f, m3 = NEG_INF, s3 = 0.f;
  for (int z = 0; z < ZCHLEN; ++z) {
    float uz = ub[z];
    float4 v = base[(size_t)z * (S_ / 4)];
    lse_acc(m0, s0, v.x + uz);
    lse_acc(m1, s1, v.y + uz);
    lse_acc(m2, s2, v.z + uz);
    lse_acc(m3, s3, v.w + uz);
  }
  int o4 = (b * ZCH + ch) * (S_ / 4) + tid;
  ((float4*)pm)[o4] = make_float4(m0, m1, m2, m3);
  ((float4*)ps)[o4] = make_float4(s0, s1, s2, s3);
}

// mode 0: t[b,s]    = lp2[b,s] - lse_z(lc0 + u)
// mode 1: araw[b,s] = lp2[b,s] - (lse_z(lc0 + u11) + t10[b,s])
__global__ void k_col_combine(const float* __restrict__ pm, const float* __restrict__ ps,
                              const float* __restrict__ lp2, const float* __restrict__ t10,
                              float* __restrict__ out, int mode) {
  int idx = blockIdx.x * 256 + threadIdx.x;   // over B*S
  int b = idx / S_;
  int s = idx - b * S_;
  float m = NEG_INF, sm = 0.0f;
  for (int ch = 0; ch < ZCH; ++ch) {
    int o = (b * ZCH + ch) * S_ + s;
    lse_merge(m, sm, pm[o], ps[o]);
  }
  float lse = (sm > 0.0f) ? m + logf(sm) : NEG_INF;
  float l2 = lp2[idx];
  out[idx] = (mode == 0) ? (l2 - lse) : (l2 - (lse + t10[idx]));
}

// accept -= max(accept); Tacc = t10 + accept
__global__ void k_accept(const float* __restrict__ araw, const float* __restrict__ t10,
                         float* __restrict__ Tacc) {
  __shared__ float red[512];
  int b = blockIdx.x, s = threadIdx.x;
  float a = araw[b * S_ + s];
  red[s] = a; __syncthreads();
  for (int o = 256; o; o >>= 1) { if (s < o) red[s] = fmaxf(red[s], red[s + o]); __syncthreads(); }
  Tacc[b * S_ + s] = t10[b * S_ + s] + a - red[0];
}

// out = lp1[z] + logaddexp(lc0 + A2[b,z] + Tacc[b,s], R[b,z] + lp2[b,s])  (in place, float4)
__device__ __forceinline__ float ladd1(float x, float y) {
  if (y == NEG_INF) return x;
  float mx = fmaxf(x, y), mn = fminf(x, y);
  return mx + log1pf(expf(mn - mx));
}

__global__ void k_final(float* __restrict__ lc, const float* __restrict__ A2,
                        const float* __restrict__ R, const float* __restrict__ Tacc,
                        const float* __restrict__ lp1, const float* __restrict__ lp2) {
  size_t q = (size_t)blockIdx.x * 256 + threadIdx.x;  // float4 index over B*N/4
  size_t base = q * 4;
  int b   = (int)(base >> 19);          // N_ = 2^19
  int rem = (int)(base & (N_ - 1));
  int z  = rem >> 9;
  int s4 = (rem & (S_ - 1)) >> 2;       // float4 index within row
  float az = A2[b * Z_ + z];
  float rz = R[b * Z_ + z];
  float l1 = lp1[z];
  float4 lcv = ((float4*)lc)[q];
  float4 tv  = ((const float4*)(Tacc + b * S_))[s4];
  float4 l2v = ((const float4*)(lp2 + b * S_))[s4];
  float4 res;
  res.x = l1 + ladd1(lcv.x + az + tv.x, rz + l2v.x);
  res.y = l1 + ladd1(lcv.y + az + tv.y, rz + l2v.y);
  res.z = l1 + ladd1(lcv.z + az + tv.z, rz + l2v.z);
  res.w = l1 + ladd1(lcv.w + az + tv.w, rz + l2v.w);
  ((float4*)lc)[q] = res;
}

// ---------------- host launcher ----------------

extern "C" void kernel_launch(void* const* d_in, const int* in_sizes, int n_in,
                              void* d_out, int out_size, void* d_ws, size_t ws_size,
                              hipStream_t stream) {
  (void)in_sizes; (void)n_in; (void)out_size; (void)ws_size;
  const float* s_logits = (const float*)d_in[0];
  const float* W1    = (const float*)d_in[1];
  const float* b1    = (const float*)d_in[2];
  const float* W2    = (const float*)d_in[3];
  const float* b2    = (const float*)d_in[4];
  const float* Wout  = (const float*)d_in[5];
  const float* bout  = (const float*)d_in[6];
  const float* prior = (const float*)d_in[7];
  float* out = (float*)d_out;
  float* ws  = (float*)d_ws;

  float* zlog = ws;               // Z_
  float* p    = zlog + Z_;        // B*S
  float* h1   = p    + B_ * S_;   // B*H1
  float* h2   = h1   + B_ * H1_;  // B*H2
  float* u    = h2   + B_ * H2_;  // B*Z
  float* t    = u    + B_ * Z_;   // B*S
  float* A2   = t    + B_ * S_;   // B*Z
  float* Rr   = A2   + B_ * Z_;   // B*Z
  float* Tacc = Rr   + B_ * Z_;   // B*S
  float* araw = Tacc + B_ * S_;   // B*S
  float* pm   = araw + B_ * S_;   // B*ZCH*S
  float* psum = pm   + B_ * ZCH * S_;

  // prologue: z log-softmax, s softmax, 2-layer sigmoid MLP
  k_zlogits<<<1, 256, 0, stream>>>(prior, zlog);
  k_softmax<<<B_, 256, 0, stream>>>(s_logits, p);
  k_dense_sigmoid<<<B_ * (H1_ / 256), 256, 0, stream>>>(p,  W1, b1, h1, S_,  H1_);
  k_dense_sigmoid<<<B_ * (H2_ / 256), 256, 0, stream>>>(h1, W2, b2, h2, H1_, H2_);

  // big fp32 WMMA GEMM -> lc0 resident in d_out (and L2)
  k_gemm_wmma<<<N_ / 128, 256, 0, stream>>>(h2, Wout, bout, out);

  // Sinkhorn on potentials only (lc0 is never rewritten)
  k_zero<<<(B_ * S_) / 256, 256, 0, stream>>>(t, B_ * S_);
  for (int it = 0; it < 10; ++it) {
    k_row<<<(B_ * Z_) / 8, 256, 0, stream>>>(out, t, zlog, nullptr, u, nullptr, 0);
    k_col_partial<<<B_ * ZCH, 128, 0, stream>>>(out, u, pm, psum);
    k_col_combine<<<(B_ * S_) / 256, 256, 0, stream>>>(pm, psum, s_logits, nullptr, t, 0);
  }

  // rejection: row step (u11), column LSE for accept, shift, keep-probs, final fuse
  k_row<<<(B_ * Z_) / 8, 256, 0, stream>>>(out, t, zlog, nullptr, u, nullptr, 0);   // u = u11
  k_col_partial<<<B_ * ZCH, 128, 0, stream>>>(out, u, pm, psum);
  k_col_combine<<<(B_ * S_) / 256, 256, 0, stream>>>(pm, psum, s_logits, t, araw, 1);
  k_accept<<<B_, 512, 0, stream>>>(araw, t, Tacc);
  k_row<<<(B_ * Z_) / 8, 256, 0, stream>>>(out, Tacc, zlog, u, A2, Rr, 1);
  k_final<<<(B_ * (N_ / 1024)), 256, 0, stream>>>(out, A2, Rr, Tacc, zlog, s_logits);
}